// GeometricProteinFeatures_14989435863163
// MI455X (gfx1250) — compile-verified
//
#include <hip/hip_runtime.h>
#include <hip/hip_bf16.h>
#include <math.h>

// Problem constants (match reference)
#define BB 4
#define NN 4096
#define KK 64
#define NUM_RBF 18

typedef __attribute__((ext_vector_type(2))) float v2f;
typedef __attribute__((ext_vector_type(8))) float v8f;

__device__ __forceinline__ float sgnf(float x) {
    return (x > 0.f ? 1.f : 0.f) - (x < 0.f ? 1.f : 0.f);
}

__device__ __forceinline__ void norm3(float& x, float& y, float& z) {
    float n = sqrtf(x * x + y * y + z * z);
    float inv = 1.f / fmaxf(n, 1e-12f);
    x *= inv; y *= inv; z *= inv;
}

__device__ __forceinline__ void cross3(float ax, float ay, float az,
                                       float bx, float by, float bz,
                                       float& cx, float& cy, float& cz) {
    cx = ay * bz - az * by;
    cy = az * bx - ax * bz;
    cz = ax * by - ay * bx;
}

// ---------------------------------------------------------------------------
// Kernel 1: per-node orientation matrices Oflat[b,n,9]
// Rows: o1 = norm(u2-u1), n2 = norm(u2 x u1), r2 = o1 x n2
// u2 = norm(ca[n]-ca[n-1]), u1 = norm(ca[n+1]-ca[n]); zero rows for
// n==0 or n>=N-2 (matches jnp.pad((1,2)) of the length N-3 array).
// ---------------------------------------------------------------------------
__global__ void onode_kernel(const float* __restrict__ coords,
                             float* __restrict__ Oflat) {
    int t = blockIdx.x * blockDim.x + threadIdx.x;   // t = b*N + n
    if (t >= BB * NN) return;
    int n = t % NN;
    float* o = Oflat + (size_t)t * 9;
    if (n < 1 || n > NN - 3) {
        #pragma unroll
        for (int i = 0; i < 9; ++i) o[i] = 0.f;
        return;
    }
    // ca[i] = coords[b, i, 1, :]
    const float* base = coords + ((size_t)(t - 1) * 4 + 1) * 3;   // ca[n-1]
    float p0x = base[0],      p0y = base[1],      p0z = base[2];
    float p1x = base[12 + 0], p1y = base[12 + 1], p1z = base[12 + 2];
    float p2x = base[24 + 0], p2y = base[24 + 1], p2z = base[24 + 2];

    float u2x = p1x - p0x, u2y = p1y - p0y, u2z = p1z - p0z;
    float u1x = p2x - p1x, u1y = p2y - p1y, u1z = p2z - p1z;
    norm3(u2x, u2y, u2z);
    norm3(u1x, u1y, u1z);

    float n2x, n2y, n2z;
    cross3(u2x, u2y, u2z, u1x, u1y, u1z, n2x, n2y, n2z);
    norm3(n2x, n2y, n2z);

    float o1x = u2x - u1x, o1y = u2y - u1y, o1z = u2z - u1z;
    norm3(o1x, o1y, o1z);

    float r2x, r2y, r2z;
    cross3(o1x, o1y, o1z, n2x, n2y, n2z, r2x, r2y, r2z);

    o[0] = o1x; o[1] = o1y; o[2] = o1z;
    o[3] = n2x; o[4] = n2y; o[5] = n2z;
    o[6] = r2x; o[7] = r2y; o[8] = r2z;
}

// ---------------------------------------------------------------------------
// Kernel 2: dihedral node features. One thread per flat backbone index f in
// [0, 3N). D[f]=0 for f==0 or f>=3N-2, else torsion from atoms f-1..f+2.
// Output: out_node[b, n, 0..2] = cos(D), [3..5] = sin(D), n=f/3.
// ---------------------------------------------------------------------------
__global__ void dihed_kernel(const float* __restrict__ coords,
                             float* __restrict__ out_node) {
    int t = blockIdx.x * blockDim.x + threadIdx.x;   // t = b*3N + f
    if (t >= BB * 3 * NN) return;
    int b = t / (3 * NN);
    int f = t % (3 * NN);
    float c = 1.f, s = 0.f;
    if (f >= 1 && f <= 3 * NN - 3) {
        // Xf[a] = coords[b, a/3, a%3, :]; atoms f-1 .. f+2 are contiguous
        // because the first 3 atoms of each residue are contiguous in memory
        // only within a residue -> index explicitly.
        float X[4][3];
        #pragma unroll
        for (int q = 0; q < 4; ++q) {
            int a = f - 1 + q;
            const float* p = coords + (((size_t)b * NN + (a / 3)) * 4 + (a % 3)) * 3;
            X[q][0] = p[0]; X[q][1] = p[1]; X[q][2] = p[2];
        }
        float u2x = X[1][0] - X[0][0], u2y = X[1][1] - X[0][1], u2z = X[1][2] - X[0][2];
        float u1x = X[2][0] - X[1][0], u1y = X[2][1] - X[1][1], u1z = X[2][2] - X[1][2];
        float u0x = X[3][0] - X[2][0], u0y = X[3][1] - X[2][1], u0z = X[3][2] - X[2][2];
        norm3(u2x, u2y, u2z);
        norm3(u1x, u1y, u1z);
        norm3(u0x, u0y, u0z);
        float n2x, n2y, n2z, n1x, n1y, n1z;
        cross3(u2x, u2y, u2z, u1x, u1y, u1z, n2x, n2y, n2z);
        norm3(n2x, n2y, n2z);
        cross3(u1x, u1y, u1z, u0x, u0y, u0z, n1x, n1y, n1z);
        norm3(n1x, n1y, n1z);
        float cosD = n2x * n1x + n2y * n1y + n2z * n1z;
        cosD = fminf(fmaxf(cosD, -1.f + 1e-7f), 1.f - 1e-7f);
        float D = sgnf(u2x * n1x + u2y * n1y + u2z * n1z) * acosf(cosD);
        c = __cosf(D);
        s = __sinf(D);
    }
    int n = f / 3, t3 = f % 3;
    float* op = out_node + ((size_t)b * NN + n) * 6;
    op[t3] = c;
    op[3 + t3] = s;
}

// ---------------------------------------------------------------------------
// Kernel 3: edge features. One wave (32 lanes) per node; 2 chunks of 32 edges.
// Per edge: 18 RBFs, dU = l2norm(Om @ dXn)  [via V_WMMA_F32_16X16X4_F32],
// quaternion of R = Om^T @ O_nb (scalar 3x3).
//
// WMMA mapping (wave32, f32 16x16x4):
//   A (16x4): VGPR0 = K {0 | 2}, VGPR1 = K {1 | 3} across lane halves,
//             M = lane%16. Rows 0..2 carry Om, rest zero.
//   B (4x16): VGPR0 = rows {0 | 2}, VGPR1 = rows {1 | 3}; column = edge.
//   D (16x16): lane k (0..15), VGPR v = row v, col k  -> lanes 0..15 hold
//             (Om @ dXn_k).xyz in c[0..2] for 16 edges per WMMA.
// ---------------------------------------------------------------------------
__global__ void edge_kernel(const float* __restrict__ coords,
                            const float* __restrict__ dists,
                            const int* __restrict__ eidx,
                            const float* __restrict__ Oflat,
                            float* __restrict__ out_edge) {
    const int wave = threadIdx.x >> 5;
    const int lane = threadIdx.x & 31;
    const int node = blockIdx.x * 8 + wave;          // grid sized exactly: B*N/8 blocks
    const int b = node / NN;
    const int n = node % NN;

    // Per-node (wave-uniform) data
    const float* Om = Oflat + (size_t)node * 9;
    float M[9];
    #pragma unroll
    for (int i = 0; i < 9; ++i) M[i] = Om[i];
    const float* cap = coords + ((size_t)node * 4 + 1) * 3;
    const float cax = cap[0], cay = cap[1], caz = cap[2];

    // A-matrix pack (loop invariant)
    const bool lo = lane < 16;
    const int m = lane & 15;
    const int k0 = lo ? 0 : 2;
    const int k1 = lo ? 1 : 3;
    v2f a;
    a[0] = (m < 3) ? M[m * 3 + k0] : 0.f;            // k0 in {0,2} < 3
    a[1] = (m < 3 && k1 < 3) ? M[m * 3 + k1] : 0.f;  // k1==3 -> pad 0

    const float inv_sigma = (float)NUM_RBF / 20.f;   // sigma = 20/18

    #pragma unroll
    for (int chunk = 0; chunk < 2; ++chunk) {
        const int e = chunk * 32 + lane;
        const size_t ebase = (size_t)node * KK + e;
        const int idx = eidx[ebase];
        const float d = dists[ebase];
        float* op = out_edge + ebase * 25;

        // ---- RBF features [0..17] ----
        #pragma unroll
        for (int j = 0; j < NUM_RBF; ++j) {
            float mu = (float)j * (20.f / 17.f);
            float z = (d - mu) * inv_sigma;
            op[j] = __expf(-z * z);
        }

        // ---- gather neighbor CA, displacement ----
        const float* nbp = coords + (((size_t)b * NN + idx) * 4 + 1) * 3;
        float dx = nbp[0] - cax;
        float dy = nbp[1] - cay;
        float dz = nbp[2] - caz;

        // ---- dU = l2norm(Om @ dXn) via WMMA, 16 edges per issue ----
        float r0x = 0.f, r0y = 0.f, r0z = 0.f;       // group 0 result (lanes 0-15)
        float r1x = 0.f, r1y = 0.f, r1z = 0.f;       // group 1 result (lanes 0-15)
        #pragma unroll
        for (int g = 0; g < 2; ++g) {
            int src = (lane & 15) + g * 16;
            float bx = __shfl(dx, src, 32);
            float by = __shfl(dy, src, 32);
            float bz = __shfl(dz, src, 32);
            v2f bv;
            bv[0] = lo ? bx : bz;                    // B rows {0 | 2}
            bv[1] = lo ? by : 0.f;                   // B rows {1 | 3(pad)}
            v8f c = {};
            c = __builtin_amdgcn_wmma_f32_16x16x4_f32(
                    false, a, false, bv, (short)0, c, false, false);
            if (g == 0) { r0x = c[0]; r0y = c[1]; r0z = c[2]; }
            else        { r1x = c[0]; r1y = c[1]; r1z = c[2]; }
        }
        // Route group-1 results (held by lanes 0-15) to lanes 16-31
        float g1x = __shfl(r1x, lane & 15, 32);
        float g1y = __shfl(r1y, lane & 15, 32);
        float g1z = __shfl(r1z, lane & 15, 32);
        float ux = lo ? r0x : g1x;
        float uy = lo ? r0y : g1y;
        float uz = lo ? r0z : g1z;
        norm3(ux, uy, uz);
        op[18] = ux; op[19] = uy; op[20] = uz;

        // ---- R = Om^T @ O_nb, quaternion [21..24] ----
        const float* Onb = Oflat + ((size_t)b * NN + idx) * 9;
        float Nb[9];
        #pragma unroll
        for (int i = 0; i < 9; ++i) Nb[i] = Onb[i];
        float R[3][3];
        #pragma unroll
        for (int i = 0; i < 3; ++i)
            #pragma unroll
            for (int l = 0; l < 3; ++l)
                R[i][l] = M[0 * 3 + i] * Nb[0 * 3 + l]
                        + M[1 * 3 + i] * Nb[1 * 3 + l]
                        + M[2 * 3 + i] * Nb[2 * 3 + l];

        float Rxx = R[0][0], Ryy = R[1][1], Rzz = R[2][2];
        float m0 = 0.5f * sqrtf(fabsf(1.f + Rxx - Ryy - Rzz));
        float m1 = 0.5f * sqrtf(fabsf(1.f - Rxx + Ryy - Rzz));
        float m2 = 0.5f * sqrtf(fabsf(1.f - Rxx - Ryy + Rzz));
        float qx = sgnf(R[2][1] - R[1][2]) * m0;
        float qy = sgnf(R[0][2] - R[2][0]) * m1;
        float qz = sgnf(R[1][0] - R[0][1]) * m2;
        float qw = 0.5f * sqrtf(fmaxf(0.f, 1.f + Rxx + Ryy + Rzz));
        float qn = sqrtf(qx * qx + qy * qy + qz * qz + qw * qw);
        float qi = 1.f / fmaxf(qn, 1e-12f);
        op[21] = qx * qi; op[22] = qy * qi; op[23] = qz * qi; op[24] = qw * qi;
    }
}

extern "C" void kernel_launch(void* const* d_in, const int* in_sizes, int n_in,
                              void* d_out, int out_size, void* d_ws, size_t ws_size,
                              hipStream_t stream) {
    const float* coords = (const float*)d_in[0];   // (B,N,4,3) f32
    const float* dists  = (const float*)d_in[1];   // (B,N,K)   f32
    const int*   eids   = (const int*)d_in[2];     // (B,N,K)   i32
    // d_in[3] = mask, unused by the reference math.

    float* out      = (float*)d_out;
    float* out_node = out;                                   // B*N*6
    float* out_edge = out + (size_t)BB * NN * 6;             // B*N*K*25
    float* Oflat    = (float*)d_ws;                          // B*N*9 floats

    {
        int total = BB * NN;
        onode_kernel<<<(total + 255) / 256, 256, 0, stream>>>(coords, Oflat);
    }
    {
        int total = BB * 3 * NN;
        dihed_kernel<<<(total + 255) / 256, 256, 0, stream>>>(coords, out_node);
    }
    {
        // 8 waves per block, 1 node per wave; B*N divisible by 8 -> full waves,
        // EXEC all-ones around WMMA.
        edge_kernel<<<(BB * NN) / 8, 256, 0, stream>>>(coords, dists, eids,
                                                       Oflat, out_edge);
    }
}